// SageBackbone_69595650065051
// MI455X (gfx1250) — compile-verified
//
#include <hip/hip_runtime.h>
#include <hip/hip_bf16.h>

typedef __attribute__((ext_vector_type(2))) float v2f;
typedef __attribute__((ext_vector_type(8))) float v8f;

#define DHID 128

// ---------------------------------------------------------------------------
// Zero-fill workspace (ws is poisoned 0xAA by the harness; we must self-init).
// ---------------------------------------------------------------------------
__global__ void sage_zero_f32(float* __restrict__ p, size_t n) {
  size_t i = (size_t)blockIdx.x * blockDim.x + threadIdx.x;
  size_t s = (size_t)gridDim.x * blockDim.x;
  for (; i < n; i += s) p[i] = 0.0f;
}

// ---------------------------------------------------------------------------
// Edge scatter: one wave32 per edge. Each lane loads 4 consecutive floats of
// the source row (coalesced float4 => 512B per wave, full row) and atomically
// adds them into the destination row (L2-resident global_atomic_add_f32).
// Lane 0 accumulates degree (layer 1 only; degree identical for both layers).
// ---------------------------------------------------------------------------
__global__ __launch_bounds__(256) void sage_scatter(
    const float* __restrict__ xin, const int* __restrict__ src,
    const int* __restrict__ dst, float* __restrict__ agg,
    float* __restrict__ deg, int E) {
  int wv = (int)((blockIdx.x * (unsigned)blockDim.x + threadIdx.x) >> 5);
  int lane = threadIdx.x & 31;
  if (wv >= E) return;
  int r = src[wv];
  int c = dst[wv];
  const float4 v = *(const float4*)(xin + (size_t)r * DHID + lane * 4);
  float* d = agg + (size_t)c * DHID + lane * 4;
  atomicAdd(d + 0, v.x);
  atomicAdd(d + 1, v.y);
  atomicAdd(d + 2, v.z);
  atomicAdd(d + 3, v.w);
  if (deg != nullptr && lane == 0) atomicAdd(deg + c, 1.0f);
}

// ---------------------------------------------------------------------------
// Fused SAGE layer: out = relu(mean @ wl + bl + xin @ wr)
// Block = 32 rows x 128 cols, 8 waves. Wave w computes TWO 16x16 output tiles
// (rows [0,16) and [16,32) of the block) at columns [16w,16w+16), sharing the
// same B-fragments -> 4 global weight loads per 4 fp32 WMMAs.
//
// A-fragment (16x4 f32, ISA 7.12.2): lane<16 -> row=lane, K={k,k+1};
//                                    lane>=16 -> row=lane-16, K={k+2,k+3}.
// B-fragment (4x16 f32): VGPR0 = rows K={k | k+2} striped across half-waves,
//                        VGPR1 = rows K={k+1 | k+3}.
// C/D (16x16 f32): VGPR i -> M=i (lanes 0-15) / M=i+8 (lanes 16-31), N=lane%16.
// ---------------------------------------------------------------------------
__global__ __launch_bounds__(256) void sage_gemm(
    const float* __restrict__ agg, const float* __restrict__ deg,
    const float* __restrict__ xin, const float* __restrict__ wl,
    const float* __restrict__ blv, const float* __restrict__ wr,
    float* __restrict__ out, int N) {
  __shared__ float sM[32][DHID];  // mean tile
  __shared__ float sX[32][DHID];  // input tile

  const int rowBase = blockIdx.x * 32;
  const int tid = threadIdx.x;
  const bool fullTile = (rowBase + 32 <= N);

  // Cooperative tile load: 2 * 32*128 floats over 256 threads -> 16+16 each.
  {
    int i = tid * 16;
    int r = i >> 7;          // 0..31
    int c = i & 127;         // multiple of 16
    int g = rowBase + r;
    int gc = (g < N) ? g : (N - 1);          // clamp (partial tiles only)
    float inv = (g < N) ? (1.0f / fmaxf(deg[gc], 1.0f)) : 0.0f;
    const float4* pa = (const float4*)(agg + (size_t)gc * DHID + c);
    const float4* px = (const float4*)(xin + (size_t)gc * DHID + c);
    float4* dm = (float4*)&sM[r][c];
    float4* dx = (float4*)&sX[r][c];
#pragma unroll
    for (int q = 0; q < 4; ++q) {
      float4 a = pa[q];
      dm[q] = make_float4(a.x * inv, a.y * inv, a.z * inv, a.w * inv);
      dx[q] = px[q];
    }
  }
  __syncthreads();

  const int wave = tid >> 5;           // 0..7 -> column tile
  const int lane = tid & 31;
  const int r = lane & 15;             // A row / B,C column within tile
  const int khalf = (lane >> 4) << 1;  // 0 for lanes 0-15, 2 for lanes 16-31
  const int c = wave * 16 + r;         // global output column

  v8f acc0 = {};
  v8f acc1 = {};
#pragma unroll 4
  for (int k = 0; k < DHID; k += 4) {
    int kk = k + khalf;
    v2f bL, bR, a0, a1;
    bL.x = wl[(size_t)kk * DHID + c];
    bL.y = wl[(size_t)(kk + 1) * DHID + c];
    bR.x = wr[(size_t)kk * DHID + c];
    bR.y = wr[(size_t)(kk + 1) * DHID + c];
    float2 t;
    t = *(const float2*)&sM[r][kk];      a0.x = t.x; a0.y = t.y;
    t = *(const float2*)&sM[16 + r][kk]; a1.x = t.x; a1.y = t.y;
    acc0 = __builtin_amdgcn_wmma_f32_16x16x4_f32(false, a0, false, bL,
                                                 (short)0, acc0, false, false);
    acc1 = __builtin_amdgcn_wmma_f32_16x16x4_f32(false, a1, false, bL,
                                                 (short)0, acc1, false, false);
    t = *(const float2*)&sX[r][kk];      a0.x = t.x; a0.y = t.y;
    t = *(const float2*)&sX[16 + r][kk]; a1.x = t.x; a1.y = t.y;
    acc0 = __builtin_amdgcn_wmma_f32_16x16x4_f32(false, a0, false, bR,
                                                 (short)0, acc0, false, false);
    acc1 = __builtin_amdgcn_wmma_f32_16x16x4_f32(false, a1, false, bR,
                                                 (short)0, acc1, false, false);
  }

  // Epilogue: bias + relu + store. Lane holds 8 rows per tile of column c.
  const float bias = blv[c];
  const int mOff = (lane >> 4) << 3;  // lanes 16-31 hold M = i + 8
  const int m0 = rowBase + mOff;
  if (fullTile) {
    // Full tile (always true for N % 32 == 0): straight-line stores,
    // 16-lane coalesced 64B per instruction.
#pragma unroll
    for (int i = 0; i < 8; ++i) {
      out[(size_t)(m0 + i) * DHID + c]      = fmaxf(acc0[i] + bias, 0.0f);
      out[(size_t)(m0 + 16 + i) * DHID + c] = fmaxf(acc1[i] + bias, 0.0f);
    }
  } else {
#pragma unroll
    for (int i = 0; i < 8; ++i) {
      int ma = m0 + i, mb = m0 + 16 + i;
      if (ma < N) out[(size_t)ma * DHID + c] = fmaxf(acc0[i] + bias, 0.0f);
      if (mb < N) out[(size_t)mb * DHID + c] = fmaxf(acc1[i] + bias, 0.0f);
    }
  }
}

// ---------------------------------------------------------------------------
// Launcher. Inputs (setup_inputs order):
//   0: x [N,128] f32   1: edge_index [2,E] i32
//   2: w1l [128,128]   3: b1l [128]   4: w1r [128,128]
//   5: w2l [128,128]   6: b2l [128]   7: w2r [128,128]
// Workspace layout: agg [N*128] | deg [N] | h [N*128]   (all f32)
// ---------------------------------------------------------------------------
extern "C" void kernel_launch(void* const* d_in, const int* in_sizes, int n_in,
                              void* d_out, int out_size, void* d_ws,
                              size_t ws_size, hipStream_t stream) {
  const float* x   = (const float*)d_in[0];
  const int*   ei  = (const int*)d_in[1];
  const float* w1l = (const float*)d_in[2];
  const float* b1l = (const float*)d_in[3];
  const float* w1r = (const float*)d_in[4];
  const float* w2l = (const float*)d_in[5];
  const float* b2l = (const float*)d_in[6];
  const float* w2r = (const float*)d_in[7];
  float* out = (float*)d_out;

  const int N = in_sizes[0] / DHID;
  const int E = in_sizes[1] / 2;
  const int* src = ei;          // edge_index[0]
  const int* dst = ei + E;      // edge_index[1]

  float* agg = (float*)d_ws;
  float* deg = agg + (size_t)N * DHID;
  float* h   = deg + N;

  const int sblocks = (E + 7) / 8;         // 8 waves (edges) per 256-thr block
  const int mtiles  = (N + 31) / 32;

  // Layer 1
  sage_zero_f32<<<2048, 256, 0, stream>>>(agg, (size_t)N * DHID + N);  // agg+deg
  sage_scatter<<<sblocks, 256, 0, stream>>>(x, src, dst, agg, deg, E);
  sage_gemm<<<mtiles, 256, 0, stream>>>(agg, deg, x, w1l, b1l, w1r, h, N);

  // Layer 2 (degree unchanged; reuse agg buffer)
  sage_zero_f32<<<2048, 256, 0, stream>>>(agg, (size_t)N * DHID);
  sage_scatter<<<sblocks, 256, 0, stream>>>(h, src, dst, agg, nullptr, E);
  sage_gemm<<<mtiles, 256, 0, stream>>>(agg, deg, h, w2l, b2l, w2r, out, N);
}